// jordon_combined_layer_83202106458326
// MI455X (gfx1250) — compile-verified
//
#include <hip/hip_runtime.h>

// CDNA5 / gfx1250: wave32, WMMA 16x16x4 FP32.
typedef __attribute__((ext_vector_type(2))) float v2f;
typedef __attribute__((ext_vector_type(8))) float v8f;

#define B_ROWS 2097152
#define DD 64
#define UU 64
#define ROWS_PER_TILE 16
#define N_TILES (B_ROWS / ROWS_PER_TILE)   // 131072

__global__ void __launch_bounds__(256)
jordon_fused_wmma_kernel(const float* __restrict__ x,
                         const float* __restrict__ state,
                         const float* __restrict__ input_W,
                         const float* __restrict__ state_W,
                         const float* __restrict__ bvec,
                         const float* __restrict__ output_W,
                         const float* __restrict__ output_b,
                         float* __restrict__ out,
                         int n_tiles) {
  const int lane  = threadIdx.x & 31;
  const int half  = lane >> 4;   // 0: K even-pair / rows r, 1: K odd-pair / rows r+8
  const int l15   = lane & 15;

  const int waves_per_block = blockDim.x >> 5;
  const int gwave   = blockIdx.x * waves_per_block + (threadIdx.x >> 5);
  const int n_waves = gridDim.x * waves_per_block;

  // ---------------- Wave-invariant preload ----------------
  // B-fragment (4x16 KxN tile of input_W) for step t, N-tile j:
  //   lane: u = j*16 + l15 ; k = 4t + 2*half ; holds {W[k][u], W[k+1][u]}
  v2f bfrag[4][16];
#pragma unroll
  for (int j = 0; j < 4; ++j) {
    const int u = j * 16 + l15;
#pragma unroll
    for (int t = 0; t < 16; ++t) {
      const int k = 4 * t + 2 * half;
      v2f v;
      v.x = input_W[k * UU + u];
      v.y = input_W[(k + 1) * UU + u];
      bfrag[j][t] = v;
    }
  }

  // Per-lane bias/state term and output weight for column u = j*16 + l15
  // (C/D layout: VGPR r, lane n -> D[M = r + 8*half][N = n], N indexed by l15)
  const float st = state[0];
  float sb[4], ow[4];
#pragma unroll
  for (int j = 0; j < 4; ++j) {
    const int u = j * 16 + l15;
    sb[j] = bvec[u] + st * state_W[u];
    ow[j] = output_W[u];
  }
  const float ob = output_b[0];

  // ---------------- Persistent grid-stride over 16-row tiles ----------------
  for (int tile = gwave; tile < n_tiles; tile += n_waves) {
    const int row_base = tile * ROWS_PER_TILE;

    // Prefetch next tile's x block (speculative; dropped if OOB/invalid)
    if (tile + n_waves < n_tiles) {
      __builtin_prefetch(x + (size_t)(tile + n_waves) * ROWS_PER_TILE * DD, 0, 3);
    }

    // A-fragments: lane reads row (row_base + l15), cols 4t + 2*half .. +1
    const float* xrow = x + (size_t)(row_base + l15) * DD + 2 * half;
    v2f afrag[16];
#pragma unroll
    for (int t = 0; t < 16; ++t) {
      afrag[t] = *(const v2f*)(xrow + 4 * t);
    }

    float res[8];
#pragma unroll
    for (int r = 0; r < 8; ++r) res[r] = 0.0f;

#pragma unroll
    for (int j = 0; j < 4; ++j) {
      v8f acc = {};
#pragma unroll
      for (int t = 0; t < 16; ++t) {
        // D = A(16x4) x B(4x16) + C, FP32
        acc = __builtin_amdgcn_wmma_f32_16x16x4_f32(
            /*neg_a=*/false, afrag[t],
            /*neg_b=*/false, bfrag[j][t],
            /*c_mod=*/(short)0, acc,
            /*reuse_a=*/false, /*reuse_b=*/false);
      }
      // Fused epilogue: sigmoid(acc + b[u] + s[u]) * output_W[u]
#pragma unroll
      for (int r = 0; r < 8; ++r) {
        const float h = 1.0f / (1.0f + __expf(-(acc[r] + sb[j])));
        res[r] += h * ow[j];
      }
    }

    // Horizontal reduce over the 16 N-lanes of each half-wave.
    // After butterfly: lane 0 = row (row_base + r), lane 16 = row (row_base + r + 8).
#pragma unroll
    for (int r = 0; r < 8; ++r) {
      float v = res[r];
      v += __shfl_xor(v, 1, 32);
      v += __shfl_xor(v, 2, 32);
      v += __shfl_xor(v, 4, 32);
      v += __shfl_xor(v, 8, 32);
      if (l15 == 0) {
        out[row_base + r + half * 8] = v + ob;
      }
    }
  }
}

extern "C" void kernel_launch(void* const* d_in, const int* in_sizes, int n_in,
                              void* d_out, int out_size, void* d_ws, size_t ws_size,
                              hipStream_t stream) {
  const float* x        = (const float*)d_in[0];  // [B, 64]
  const float* state    = (const float*)d_in[1];  // [1, 1]
  const float* input_W  = (const float*)d_in[2];  // [64, 64]
  const float* state_W  = (const float*)d_in[3];  // [1, 1, 64]
  const float* bvec     = (const float*)d_in[4];  // [1, 64]
  const float* output_W = (const float*)d_in[5];  // [64, 1]
  const float* output_b = (const float*)d_in[6];  // [1]
  float* out            = (float*)d_out;          // [B, 1]

  // 1024 blocks x 256 threads = 8192 waves; 131072 tiles -> 16 tiles/wave,
  // amortizing the 16KB weight-fragment preload.
  dim3 grid(1024), block(256);
  jordon_fused_wmma_kernel<<<grid, block, 0, stream>>>(
      x, state, input_W, state_W, bvec, output_W, output_b, out, N_TILES);
}